// MultiAgentWorldModel_10599979286565
// MI455X (gfx1250) — compile-verified
//
#include <hip/hip_runtime.h>
#include <hip/hip_bf16.h>
#include <stdint.h>

// Problem constants (match reference)
#define BB   32
#define NN   4096
#define HH   512
#define WW   512
// grid cell = float4 {prey, pred, energy, alive_flag}; feed_map read directly.
// grid bytes = 32*512*512*16 = 134,217,728 (lives in d_ws; fits MI455X 192MB L2)

typedef __attribute__((ext_vector_type(4))) unsigned int v4u_t;
typedef __attribute__((ext_vector_type(8))) int          v8i_t;
typedef __attribute__((ext_vector_type(4))) int          v4i_t;

// ---------------------------------------------------------------------------
// Kernel 1: zero the 134MB grid. 2048 blocks, 64KB chunk each.
// CDNA5 path: zero 64KB of LDS once (ds_store_b128), then one
// TENSOR_STORE_FROM_LDS (TDM DMA) blasts it to global; tracked by TENSORcnt.
// ---------------------------------------------------------------------------
__global__ __launch_bounds__(256) void zero_grid_kernel(float4* __restrict__ grid) {
#if defined(__HIP_DEVICE_COMPILE__) && \
    __has_builtin(__builtin_amdgcn_tensor_store_from_lds) && \
    __has_builtin(__builtin_amdgcn_s_wait_tensorcnt)
    __shared__ uint4 lbuf[4096];                           // 64 KB of zeros
    const uint4 z4 = make_uint4(0u, 0u, 0u, 0u);
    for (int k = threadIdx.x; k < 4096; k += 256) lbuf[k] = z4;
    __syncthreads();
    // LDS byte offset of lbuf: generic shared pointers carry the LDS offset
    // in the low 32 bits (ISA 10.2).
    const unsigned lds_off = (unsigned)(uintptr_t)(void*)&lbuf[0];
    // Compiler barrier: the TDM intrinsic reads LDS out-of-band; keep the
    // zero-stores alive and ordered before the DMA issue.
    asm volatile("" ::: "memory");
    if (threadIdx.x < 32) {                                // wave 0 issues the TDM op
        unsigned long long gaddr = (unsigned long long)(uintptr_t)grid
                                 + (unsigned long long)blockIdx.x * 65536ull;
        // ---- D# group 0 (128b): count=1 | lds_addr | global_addr | type=2
        v4u_t g0;
        g0[0] = 1u;                                        // count=1, user mode
        g0[1] = lds_off;                                   // lds_addr (bytes)
        g0[2] = (unsigned)gaddr;                           // global_addr[31:0]
        g0[3] = (unsigned)((gaddr >> 32) & 0x01FFFFFFull)  // global_addr[56:32]
              | (2u << 30);                                // type = 2 ("image")
        // ---- D# group 1 (256b): data_size=8B, 1 row of 8192 elems (64KB)
        v8i_t g1;
        g1[0] = (3 << 16);            // workgroup_mask=0, data_size=3 (8 bytes)
        g1[1] = (int)(8192u << 16);   // tensor_dim0[15:0] in bits[63:48]
        g1[2] = (1 << 16);            // tensor_dim0[31:16]=0, tensor_dim1=1
        g1[3] = (int)(8192u << 16);   // tile_dim0 = 8192 in bits[127:112]
        g1[4] = 1;                    // tile_dim1 = 1, tile_dim2 = 0
        g1[5] = 8192;                 // tensor_dim0_stride[31:0]
        g1[6] = 0;                    // stride hi / dim1_stride lo
        g1[7] = 0;
        v4i_t g2 = {0, 0, 0, 0};      // <=2D tensor: groups 2/3 disabled
        v4i_t g3 = {0, 0, 0, 0};
#if defined(__clang_major__) && __clang_major__ >= 23
        v8i_t g4 = {0, 0, 0, 0, 0, 0, 0, 0};
        __builtin_amdgcn_tensor_store_from_lds(g0, g1, g2, g3, g4, 0);
#else
        __builtin_amdgcn_tensor_store_from_lds(g0, g1, g2, g3, 0);
#endif
        __builtin_amdgcn_s_wait_tensorcnt(0);
    }
#else
    // Fallback: plain B128 stores, same 64KB chunk per block
    const int base = blockIdx.x * 4096;
    const float4 z = make_float4(0.f, 0.f, 0.f, 0.f);
    for (int k = threadIdx.x; k < 4096; k += 256) grid[base + k] = z;
#endif
}

// ---------------------------------------------------------------------------
// Kernel 2: scatter. One thread per agent; live agents do ONE global_store_b128
// (prey, pred, energy, 1.0) into their cell. ~2MB of scattered writes.
// ---------------------------------------------------------------------------
__global__ __launch_bounds__(256) void scatter_kernel(
    const float*  __restrict__ e,
    const float2* __restrict__ species,
    const float*  __restrict__ alive,
    const int2*   __restrict__ pos,
    float4*       __restrict__ grid) {
    const int t = blockIdx.x * 256 + threadIdx.x;          // exactly B*N threads
    const float a = alive[t];
    if (a > 0.f) {
        const int2   p  = pos[t];
        const float2 sp = species[t];
        const float  ev = e[t];
        const int b    = t >> 12;                          // N = 4096 = 2^12
        const int cell = (b << 18) + (p.x << 9) + p.y;     // H*W=2^18, W=2^9
        grid[cell] = make_float4(sp.x, sp.y, ev, 1.0f);
    }
}

// ---------------------------------------------------------------------------
// Kernel 3: gather. One thread per output element (B,N,C,P,P) -> fully
// coalesced 128B/wave output stores; grid reads hit L2 (grid is L2-resident).
// Zero padding handled by the bounds check; c==3 reads feed_map input.
// ---------------------------------------------------------------------------
__global__ __launch_bounds__(256) void gather_kernel(
    const float* __restrict__ feed,
    const float* __restrict__ alive,
    const int2*  __restrict__ pos,
    const float* __restrict__ gridf,                       // grid as flat floats
    float*       __restrict__ out) {
    const unsigned t = blockIdx.x * 256u + threadIdx.x;    // exactly 16,384,000
    unsigned r = t;
    const unsigned j = r % 5u; r /= 5u;
    const unsigned i = r % 5u; r /= 5u;
    const unsigned c = r % 5u; r /= 5u;                    // r = agent = b*N + n
    const unsigned agent = r;
    const unsigned b = agent >> 12;

    const float a = alive[agent];
    float v = 0.f;
    if (a > 0.f) {
        const int2 p = pos[agent];
        const int x = p.x - 2 + (int)i;
        const int y = p.y - 2 + (int)j;
        if ((unsigned)x < (unsigned)HH && (unsigned)y < (unsigned)WW) {
            const unsigned cell = (b << 18) + ((unsigned)x << 9) + (unsigned)y;
            v = (c == 3u) ? feed[cell]
                          : gridf[(cell << 2) + (c == 4u ? 3u : c)];
        }
    }
    out[t] = v * a;                                        // alive scaling
}

// ---------------------------------------------------------------------------
extern "C" void kernel_launch(void* const* d_in, const int* in_sizes, int n_in,
                              void* d_out, int out_size, void* d_ws, size_t ws_size,
                              hipStream_t stream) {
    // setup_inputs order: e, species, feed_map, alive, pos
    const float*  e       = (const float*) d_in[0];        // (B,N,1)
    const float2* species = (const float2*)d_in[1];        // (B,N,2)
    const float*  feed    = (const float*) d_in[2];        // (B,H,W)
    const float*  alive   = (const float*) d_in[3];        // (B,N,1)
    const int2*   pos     = (const int2*)  d_in[4];        // (B,N,2) int32
    float*        out     = (float*)d_out;                 // (B,N,5,5,5)

    float4* grid = (float4*)d_ws;                          // 134,217,728 bytes
    (void)in_sizes; (void)n_in; (void)out_size; (void)ws_size;

    // 1) zero grid: 2048 blocks x 64KB (TDM tensor_store_from_lds)
    zero_grid_kernel<<<2048, 256, 0, stream>>>(grid);
    // 2) scatter live agents: B*N = 131072 threads
    scatter_kernel<<<(BB * NN) / 256, 256, 0, stream>>>(e, species, alive, pos, grid);
    // 3) gather patches: 16,384,000 threads
    gather_kernel<<<(BB * NN * 125) / 256, 256, 0, stream>>>(
        feed, alive, pos, (const float*)grid, out);
}